// NHGATModelVAE_21749714387351
// MI455X (gfx1250) — compile-verified
//
#include <hip/hip_runtime.h>
#include <hip/hip_bf16.h>

// ---------------------------------------------------------------------------
// MI455X (gfx1250, wave32) implementation of the NHGAT-VAE forward pass.
// All matmuls run on V_WMMA_F32_16X16X4_F32 (full fp32: workload is
// HBM-bandwidth bound at ~0.5 GB vs 14 GFLOP, so fp32 WMMA is free accuracy).
// GEMM is templated on tile count / transB / activation so inner loops fully
// unroll: static acc indexing -> no v_movrel around the WMMAs.
// ---------------------------------------------------------------------------

typedef __attribute__((ext_vector_type(2))) float v2f;
typedef __attribute__((ext_vector_type(8))) float v8f;

#define ALPHA 0.2f

__device__ __forceinline__ v8f wmma_f32(v2f a, v2f b, v8f c) {
  // D = A(16x4) * B(4x16) + C(16x16), fp32 throughout.
  return __builtin_amdgcn_wmma_f32_16x16x4_f32(
      /*neg_a=*/false, a, /*neg_b=*/false, b,
      /*c_mod=*/(short)0, c, /*reuse_a=*/false, /*reuse_b=*/false);
}

// ---------------------------------------------------------------------------
// Wave32 WMMA GEMM: C[M, NT*16*gridDim.y] = act(A[M,K] @ op(B))   row-major.
// One wave per 16 x (NT*16) tile of C. A tile staged via LDS so global reads
// of A are coalesced 128B lines (matters for adj, lda=4096).
// Requires M%16==0, K%4==0; grid.y covers the N dimension exactly.
// ---------------------------------------------------------------------------
template <int NT, int TRANSB, int ACT>
__global__ void gemm16_t(const float* __restrict__ A, const float* __restrict__ B,
                         float* __restrict__ C,
                         int K, int lda, int ldb, int ldc,
                         long long strideA, long long strideB, long long strideC) {
  A += (size_t)blockIdx.z * strideA;
  B += (size_t)blockIdx.z * strideB;
  C += (size_t)blockIdx.z * strideC;

  __shared__ float At[16][33];  // +1 pad: conflict-free column reads

  const int i0 = blockIdx.x * 16;
  const int n0 = blockIdx.y * (NT * 16);

  const int l = threadIdx.x;
  const int row = l & 15;   // C column lane / A row within tile
  const int half = l >> 4;  // selects K-slots {0,1} vs {2,3} in fragments

  v8f acc[NT];
#pragma unroll
  for (int t = 0; t < NT; ++t)
#pragma unroll
    for (int r = 0; r < 8; ++r) acc[t][r] = 0.0f;

  for (int k0 = 0; k0 < K; k0 += 32) {
    int kw = K - k0; if (kw > 32) kw = 32;
    if (l < kw) {  // coalesced stage of A[i0:i0+16, k0:k0+kw]
#pragma unroll
      for (int r = 0; r < 16; ++r)
        At[r][l] = A[(size_t)(i0 + r) * lda + k0 + l];
    }
    for (int k = 0; k < kw; k += 4) {
      const int c0 = k + half * 2;  // ISA 16x4 f32 A layout: lanes>=16 hold K=2,3
      v2f a; a.x = At[row][c0]; a.y = At[row][c0 + 1];
      const int kr = k0 + c0;
#pragma unroll
      for (int t = 0; t < NT; ++t) {
        const int col = n0 + t * 16 + row;
        v2f b;
        if (TRANSB) { b.x = B[(size_t)col * ldb + kr]; b.y = B[(size_t)col * ldb + kr + 1]; }
        else        { b.x = B[(size_t)kr * ldb + col]; b.y = B[(size_t)(kr + 1) * ldb + col]; }
        acc[t] = wmma_f32(a, b, acc[t]);
      }
    }
  }
  // 16x16 f32 D layout: VGPR r, lanes<16 -> row r, lanes>=16 -> row r+8, col=l%16
#pragma unroll
  for (int t = 0; t < NT; ++t)
#pragma unroll
    for (int r = 0; r < 8; ++r) {
      float v = acc[t][r];
      if (ACT == 1) v = fmaxf(v, 0.0f);
      C[(size_t)(i0 + r + half * 8) * ldc + n0 + t * 16 + row] = v;
    }
}

// ---------------------------------------------------------------------------
// e1[h,i] = Wh[h,i,:] . a[h,0:64] ; e2[h,i] = Wh[h,i,:] . a[h,64:128]
// ---------------------------------------------------------------------------
__global__ void e_kernel(const float* __restrict__ Wh, const float* __restrict__ a,
                         float* __restrict__ e1, float* __restrict__ e2,
                         int nh, int N) {
  int idx = blockIdx.x * blockDim.x + threadIdx.x;
  if (idx >= nh * N) return;
  int h = idx / N, i = idx - h * N;
  const float* w  = Wh + ((size_t)h * N + i) * 64;
  const float* av = a + h * 128;
  float s1 = 0.f, s2 = 0.f;
#pragma unroll 8
  for (int f = 0; f < 64; ++f) { s1 += w[f] * av[f]; s2 += w[f] * av[64 + f]; }
  e1[idx] = s1; e2[idx] = s2;
}

// ---------------------------------------------------------------------------
// Online softmax stats per adjacency row (masked leaky-relu scores).
// One wave per row; all nh heads handled in a single pass over adj.
// ---------------------------------------------------------------------------
__global__ void gat_stats(const float* __restrict__ adj,
                          const float* __restrict__ e1, const float* __restrict__ e2,
                          float* __restrict__ mrow, float* __restrict__ srow,
                          int nh, int N) {
  const int i = blockIdx.x;
  const int l = threadIdx.x;
  float mx[4], sm[4], e1v[4];
#pragma unroll
  for (int h = 0; h < 4; ++h) { mx[h] = -1e30f; sm[h] = 0.f; e1v[h] = 0.f; }
  for (int h = 0; h < nh; ++h) e1v[h] = e1[h * N + i];
  const float* arow = adj + (size_t)i * N;
  for (int j = l; j < N; j += 32) {
    float aij = arow[j];
    if (aij > 0.f) {
      for (int h = 0; h < nh; ++h) {
        float e = e1v[h] + e2[h * N + j];
        e = e > 0.f ? e : ALPHA * e;
        float nm = fmaxf(mx[h], e);
        sm[h] = sm[h] * __expf(mx[h] - nm) + __expf(e - nm);
        mx[h] = nm;
      }
    }
  }
  for (int off = 16; off > 0; off >>= 1) {
    for (int h = 0; h < nh; ++h) {
      float om = __shfl_xor(mx[h], off, 32);
      float os = __shfl_xor(sm[h], off, 32);
      float nm = fmaxf(mx[h], om);
      sm[h] = sm[h] * __expf(mx[h] - nm) + os * __expf(om - nm);
      mx[h] = nm;
    }
  }
  if (l == 0)
    for (int h = 0; h < nh; ++h) { mrow[h * N + i] = mx[h]; srow[h * N + i] = sm[h]; }
}

// ---------------------------------------------------------------------------
// out[i, h*64 + c] = elu( softmax_row(att) @ Wh[h] )   for a 16-row tile.
// Attention values are generated on the fly directly into WMMA A-fragments
// (never materialized): lane l covers row l%16, K-slots (l/16)*2 .. +1.
// adj tile staged through LDS for coalescing; Wh fragments served by L2/WGP$.
// ---------------------------------------------------------------------------
__device__ __forceinline__ float attv(float aij, float e, float mr, float sinv) {
  if (aij <= 0.f) return 0.f;  // masked: exp(NEG - m)/s == 0
  e = e > 0.f ? e : ALPHA * e; // leaky-relu on unmasked scores
  return __expf(e - mr) * sinv;
}

__global__ void gat_aggregate(const float* __restrict__ adj,
                              const float* __restrict__ Wh,   // [nh, N, 64]
                              const float* __restrict__ e1, const float* __restrict__ e2,
                              const float* __restrict__ mrow, const float* __restrict__ srow,
                              float* __restrict__ outp,       // [N, outLd]
                              int outLd, int N) {
  __shared__ float adjt[16][33];
  __shared__ float e2s[32];

  const int i0 = blockIdx.x * 16;
  const int h  = blockIdx.y;
  const int l  = threadIdx.x;
  const int row = l & 15, half = l >> 4;

  const float* WhH = Wh + (size_t)h * N * 64;
  const float e1r  = e1[h * N + i0 + row];
  const float mr   = mrow[h * N + i0 + row];
  const float sinv = 1.0f / srow[h * N + i0 + row];

  v8f acc[4];
#pragma unroll
  for (int t = 0; t < 4; ++t)
#pragma unroll
    for (int r = 0; r < 8; ++r) acc[t][r] = 0.0f;

  for (int j0 = 0; j0 < N; j0 += 32) {
#pragma unroll
    for (int r = 0; r < 16; ++r)  // coalesced 128B lines
      adjt[r][l] = adj[(size_t)(i0 + r) * N + j0 + l];
    e2s[l] = e2[h * N + j0 + l];
#pragma unroll
    for (int k = 0; k < 32; k += 4) {
      const int c0 = k + half * 2;
      v2f a;
      a.x = attv(adjt[row][c0],     e1r + e2s[c0],     mr, sinv);
      a.y = attv(adjt[row][c0 + 1], e1r + e2s[c0 + 1], mr, sinv);
      const float* Bb = WhH + (size_t)(j0 + c0) * 64;
#pragma unroll
      for (int t = 0; t < 4; ++t) {
        v2f b; b.x = Bb[t * 16 + row]; b.y = Bb[64 + t * 16 + row];
        acc[t] = wmma_f32(a, b, acc[t]);
      }
    }
  }
#pragma unroll
  for (int t = 0; t < 4; ++t)
#pragma unroll
    for (int r = 0; r < 8; ++r) {
      float v = acc[t][r];
      v = v > 0.f ? v : (__expf(v) - 1.0f);  // ELU
      outp[(size_t)(i0 + r + half * 8) * outLd + h * 64 + t * 16 + row] = v;
    }
}

// ---------------------------------------------------------------------------
// z = eps * exp(logvar) + mu   (mu/logvar already live in d_out's tail)
// ---------------------------------------------------------------------------
__global__ void z_kernel(const float* __restrict__ eps, const float* __restrict__ mu,
                         const float* __restrict__ logvar, float* __restrict__ z, int n) {
  int i = blockIdx.x * blockDim.x + threadIdx.x;
  if (i < n) z[i] = eps[i] * __expf(logvar[i]) + mu[i];
}

// ---------------------------------------------------------------------------
extern "C" void kernel_launch(void* const* d_in, const int* in_sizes, int n_in,
                              void* d_out, int out_size, void* d_ws, size_t ws_size,
                              hipStream_t stream) {
  constexpr int N = 4096, D = 512, F1 = 64, F2 = 32, F3 = 16, H = 4;

  const float* x       = (const float*)d_in[0];
  const float* adj     = (const float*)d_in[1];
  const float* W_heads = (const float*)d_in[2];  // [H,512,64]
  const float* a_heads = (const float*)d_in[3];  // [H,128,1]
  const float* W_att   = (const float*)d_in[4];  // [256,64]
  const float* a_att   = (const float*)d_in[5];  // [128,1]
  const float* W1      = (const float*)d_in[6];  // [64,32]
  const float* W2      = (const float*)d_in[7];  // [32,16]
  const float* W3      = (const float*)d_in[8];  // [32,16]
  const float* eps     = (const float*)d_in[9];  // [N,16]

  float* out = (float*)d_out;  // adj_rec | mu | logvar
  float* mu     = out + (size_t)N * N;
  float* logvar = mu + (size_t)N * F3;

  float* ws = (float*)d_ws;
  float* Wh_heads = ws; ws += (size_t)H * N * F1;  // [4,4096,64]
  float* Wh_att   = ws; ws += (size_t)N * F1;
  float* cat      = ws; ws += (size_t)N * H * F1;  // [4096,256]
  float* gc       = ws; ws += (size_t)N * F1;
  float* e1h = ws; ws += (size_t)H * N;
  float* e2h = ws; ws += (size_t)H * N;
  float* mh  = ws; ws += (size_t)H * N;
  float* sh  = ws; ws += (size_t)H * N;
  float* e1a = ws; ws += N;
  float* e2a = ws; ws += N;
  float* ma  = ws; ws += N;
  float* sa  = ws; ws += N;
  float* t1  = ws; ws += (size_t)N * F2;
  float* h1  = ws; ws += (size_t)N * F2;
  float* t2  = ws; ws += (size_t)N * F3;
  float* t3  = ws; ws += (size_t)N * F3;
  float* zb  = ws; ws += (size_t)N * F3;

  dim3 w32(32);

  // 1) Wh_heads[h] = x @ W_heads[h]  (batched over heads via grid.z)
  gemm16_t<4, 0, 0><<<dim3(N / 16, 1, H), w32, 0, stream>>>(
      x, W_heads, Wh_heads, D, D, F1, F1,
      0LL, (long long)D * F1, (long long)N * F1);
  // 2) per-node attention pre-scores
  e_kernel<<<dim3((H * N + 127) / 128), dim3(128), 0, stream>>>(Wh_heads, a_heads, e1h, e2h, H, N);
  // 3) masked-softmax row stats (all heads, one adj pass)
  gat_stats<<<dim3(N), w32, 0, stream>>>(adj, e1h, e2h, mh, sh, H, N);
  // 4) heads aggregation + ELU, concat into cat[N, 256]
  gat_aggregate<<<dim3(N / 16, H), w32, 0, stream>>>(adj, Wh_heads, e1h, e2h, mh, sh, cat, H * F1, N);

  // 5) attention-GAT stage
  gemm16_t<4, 0, 0><<<dim3(N / 16, 1, 1), w32, 0, stream>>>(
      cat, W_att, Wh_att, H * F1, H * F1, F1, F1, 0LL, 0LL, 0LL);
  e_kernel<<<dim3((N + 127) / 128), dim3(128), 0, stream>>>(Wh_att, a_att, e1a, e2a, 1, N);
  gat_stats<<<dim3(N), w32, 0, stream>>>(adj, e1a, e2a, ma, sa, 1, N);
  gat_aggregate<<<dim3(N / 16, 1), w32, 0, stream>>>(adj, Wh_att, e1a, e2a, ma, sa, gc, F1, N);

  // 6) GCN layers: h1 = relu(adj @ (gc @ W1))
  gemm16_t<2, 0, 0><<<dim3(N / 16, 1, 1), w32, 0, stream>>>(gc, W1, t1, F1, F1, F2, F2, 0LL, 0LL, 0LL);
  gemm16_t<2, 0, 1><<<dim3(N / 16, 1, 1), w32, 0, stream>>>(adj, t1, h1, N, N, F2, F2, 0LL, 0LL, 0LL);
  // mu = adj @ (h1 @ W2), logvar = adj @ (h1 @ W3) -> straight into d_out tail
  gemm16_t<1, 0, 0><<<dim3(N / 16, 1, 1), w32, 0, stream>>>(h1, W2, t2, F2, F2, F3, F3, 0LL, 0LL, 0LL);
  gemm16_t<1, 0, 0><<<dim3(N / 16, 1, 1), w32, 0, stream>>>(h1, W3, t3, F2, F2, F3, F3, 0LL, 0LL, 0LL);
  gemm16_t<1, 0, 0><<<dim3(N / 16, 1, 1), w32, 0, stream>>>(adj, t2, mu,     N, N, F3, F3, 0LL, 0LL, 0LL);
  gemm16_t<1, 0, 0><<<dim3(N / 16, 1, 1), w32, 0, stream>>>(adj, t3, logvar, N, N, F3, F3, 0LL, 0LL, 0LL);

  // 7) reparameterize + inner-product decoder: adj_rec = z @ z.T
  z_kernel<<<dim3((N * F3 + 255) / 256), dim3(256), 0, stream>>>(eps, mu, logvar, zb, N * F3);
  gemm16_t<4, 1, 0><<<dim3(N / 16, N / 64, 1), w32, 0, stream>>>(
      zb, zb, out, F3, F3, F3, N, 0LL, 0LL, 0LL);
}